// NeuralMemory_76124000354878
// MI455X (gfx1250) — compile-verified
//
#include <hip/hip_runtime.h>

typedef float v2f __attribute__((ext_vector_type(2)));
typedef float v8f __attribute__((ext_vector_type(8)));

#define ALPHA_ 0.999f
#define ETA_   0.6f
#define THETA_ 0.05f
#define B_  8
#define T_  1024
#define E_  64
#define TM_ 128       // token chunk per block
#define NSPLIT_ 4     // T-scan splits per batch in the update kernel

// ---------------- WMMA helpers (f32 16x16x4, ISA 7.12.2 layouts) ----------------

__device__ __forceinline__ v8f wmma4(v2f a, v2f b, v8f c) {
  // (neg_a, A, neg_b, B, c_mod, C, reuse_a, reuse_b)
  return __builtin_amdgcn_wmma_f32_16x16x4_f32(false, a, false, b, (short)0, c,
                                               false, false);
}

// A tile (16x4): rows row0..row0+15, k cols k0..k0+3 of row-major src (ld floats)
__device__ __forceinline__ v2f ldA(const float* src, int ld, int row0, int k0, int lane) {
  int m = lane & 15, kk = (lane >> 4) * 2;
  const float* p = src + (row0 + m) * ld + k0 + kk;
  v2f a; a.x = p[0]; a.y = p[1]; return a;
}
// A tile where logical A = src^T : A[m][kk] = src[(k0+kk)*ld + m0+m]
__device__ __forceinline__ v2f ldAt(const float* src, int ld, int m0, int k0, int lane) {
  int m = lane & 15, kk = (lane >> 4) * 2;
  v2f a; a.x = src[(k0 + kk) * ld + m0 + m]; a.y = src[(k0 + kk + 1) * ld + m0 + m];
  return a;
}
// B tile for C = X @ W^T : B[kk][n] = W[(n0+n)*ld + k0+kk]
__device__ __forceinline__ v2f ldBt(const float* W, int ld, int n0, int k0, int lane) {
  int n = lane & 15, kk = (lane >> 4) * 2;
  const float* p = W + (n0 + n) * ld + k0 + kk;
  v2f b; b.x = p[0]; b.y = p[1]; return b;
}
// B tile for C = X @ W : B[kk][n] = W[(k0+kk)*ld + n0+n]
__device__ __forceinline__ v2f ldBn(const float* W, int ld, int n0, int k0, int lane) {
  int n = lane & 15, kk = (lane >> 4) * 2;
  v2f b; b.x = W[(k0 + kk) * ld + n0 + n]; b.y = W[(k0 + kk + 1) * ld + n0 + n];
  return b;
}

// 16-row strip (K=64) x 64-col GEMM, 4 independent accumulator chains, with
// double-buffered B fragments so next-k loads stay in flight behind the WMMAs.
// BT=true -> C = A @ W^T (ldBt), BT=false -> C = A @ W (ldBn)
template <bool BT>
__device__ __forceinline__ void gemm16x64(const v2f af[16], const float* W, int ld,
                                          int lane, v8f acc[4]) {
  v2f bt[4], bn[4];
#pragma unroll
  for (int j = 0; j < 4; ++j)
    bt[j] = BT ? ldBt(W, ld, j * 16, 0, lane) : ldBn(W, ld, j * 16, 0, lane);
#pragma unroll
  for (int k = 0; k < 16; ++k) {
    if (k < 15) {
#pragma unroll
      for (int j = 0; j < 4; ++j)
        bn[j] = BT ? ldBt(W, ld, j * 16, (k + 1) * 4, lane)
                   : ldBn(W, ld, j * 16, (k + 1) * 4, lane);
    }
#pragma unroll
    for (int j = 0; j < 4; ++j) acc[j] = wmma4(af[k], bt[j], acc[j]);
#pragma unroll
    for (int j = 0; j < 4; ++j) bt[j] = bn[j];
  }
}

__device__ __forceinline__ float fast_sig(float z) {
  // v_exp_f32 + v_add + v_rcp_f32 (1-ULP) — avoids the IEEE div expansion
  return __builtin_amdgcn_rcpf(1.f + __expf(-z));
}
__device__ __forceinline__ float silu_f(float z)  { return z * fast_sig(z); }
__device__ __forceinline__ float silu_df(float z) {
  float s = fast_sig(z); return s * (1.f + z * (1.f - s));
}

// ---------------- kernel 1: silu(x @ {K,V,Q}w^T) -> keyA|valA|qA ----------------

__global__ void nm_proj(const float* __restrict__ x,
                        const float* __restrict__ Kw,
                        const float* __restrict__ Qw,
                        const float* __restrict__ Vw,
                        float* __restrict__ keyA,  // [B,T,E] (valA,qA follow contiguously)
                        float* __restrict__ valA,
                        float* __restrict__ qA) {
  extern __shared__ float sX[];  // TM_*E_
  const int nTb = T_ / TM_;
  const int b = blockIdx.x / nTb;
  const int t0 = (blockIdx.x % nTb) * TM_;
  const int tid = threadIdx.x;

  const float* xsrc = x + (size_t)(b * T_ + t0) * E_;
  for (int i = tid; i < TM_ * E_ / 4; i += blockDim.x)
    ((float4*)sX)[i] = ((const float4*)xsrc)[i];
  __syncthreads();

  const int wave = tid >> 5, lane = tid & 31;
  const int row0 = wave * 16;
  const int half = lane >> 4, nn = lane & 15;

  v2f af[16];
#pragma unroll
  for (int k = 0; k < 16; ++k) af[k] = ldA(sX, E_, row0, k * 4, lane);

  const float* Ws[3] = {Kw, Vw, Qw};
  float*       Os[3] = {keyA, valA, qA};
  for (int mtx = 0; mtx < 3; ++mtx) {
    float* O = Os[mtx] + (size_t)(b * T_ + t0) * E_;
    v8f acc[4] = {};
    gemm16x64<true>(af, Ws[mtx], E_, lane, acc);
#pragma unroll
    for (int j = 0; j < 4; ++j)
#pragma unroll
      for (int r = 0; r < 8; ++r)
        O[(row0 + r + 8 * half) * E_ + j * 16 + nn] = silu_f(acc[j][r]);
  }
}

// ------------- kernel 2: deterministic per-(mtx,b,e) column sum of squares -------------

__global__ void nm_colnorm(const float* __restrict__ arrs,  // keyA base (3 arrays contiguous)
                           float* __restrict__ norms) {     // 3*B_*E_ sumsq
  const int wave = threadIdx.x >> 5, lane = threadIdx.x & 31;
  const int g = blockIdx.x * (blockDim.x >> 5) + wave;       // column id, 3*B_*E_ total
  const int mtx = g / (B_ * E_);
  const int rem = g % (B_ * E_);
  const int b = rem / E_, e = rem % E_;
  const float* a = arrs + (size_t)mtx * B_ * T_ * E_ + (size_t)b * T_ * E_ + e;
  float s = 0.f;
  for (int t = lane; t < T_; t += 32) { float v = a[(size_t)t * E_]; s += v * v; }
#pragma unroll
  for (int off = 16; off > 0; off >>= 1) s += __shfl_xor(s, off, 32);
  if (lane == 0) norms[g] = s;
}

// ------- kernel 3: fast-weight gradient accumulation (NSPLIT_ blocks per batch) -------

__global__ void nm_update(const float* __restrict__ keyA,
                          const float* __restrict__ valA,
                          const float* __restrict__ norms,
                          const float* __restrict__ W1, const float* __restrict__ b1,
                          const float* __restrict__ W2, const float* __restrict__ b2,
                          float* __restrict__ W1p, float* __restrict__ W2p,
                          float* __restrict__ b1p, float* __restrict__ b2p) {
  extern __shared__ float sm[];
  float* sK  = sm;                 // TM_*E_  normalized keys chunk
  float* sH  = sK + TM_ * E_;      // silu(hpre)
  float* sSp = sH + TM_ * E_;      // silu'(hpre), later overwritten by dhpre
  float* sDp = sSp + TM_ * E_;     // dp
  __shared__ float coeffS[T_ / NSPLIT_];
  __shared__ float invK[E_], invV[E_];

  const int b = blockIdx.x / NSPLIT_;
  const int split = blockIdx.x % NSPLIT_;
  const int tbase = split * (T_ / NSPLIT_);
  const int tid = threadIdx.x, wave = tid >> 5, lane = tid & 31;
  const int row0 = wave * 16, half = lane >> 4, nn = lane & 15;

  for (int t = tid; t < T_ / NSPLIT_; t += blockDim.x) {
    float nf = (float)(T_ - (tbase + t));  // = n+1 with n = T-1-t
    coeffS[t] = -THETA_ * (__powf(ALPHA_, nf) - __powf(ETA_, nf)) / (ALPHA_ - ETA_);
  }
  if (tid < E_) {
    invK[tid] = 1.f / fmaxf(sqrtf(norms[0 * B_ * E_ + b * E_ + tid]), 1e-12f);
    invV[tid] = 1.f / fmaxf(sqrtf(norms[1 * B_ * E_ + b * E_ + tid]), 1e-12f);
  }
  __syncthreads();

  // per-wave output-tile ownership for the 64x64 accumulations (2 tiles per wave)
  const int m0  = (wave & 3) * 16;
  const int n0a = (wave >> 2) * 32;
  v8f w1a0 = {}, w1a1 = {}, w2a0 = {}, w2a1 = {};
  float rb = 0.f;  // bias accumulator (waves 0-1 -> b1 cols, waves 2-3 -> b2 cols)

  for (int tc = 0; tc < T_ / NSPLIT_ / TM_; ++tc) {
    const int t0 = tc * TM_;  // local within this split
    // 1) normalized keys chunk -> LDS
    const float* ksrc = keyA + (size_t)(b * T_ + tbase + t0) * E_;
    for (int i = tid; i < TM_ * E_; i += blockDim.x) sK[i] = ksrc[i] * invK[i & (E_ - 1)];
    __syncthreads();

    // 2) Hpre = sK @ W1^T + b1 ; store silu and silu'
    {
      v2f af[16];
#pragma unroll
      for (int k = 0; k < 16; ++k) af[k] = ldA(sK, E_, row0, k * 4, lane);
      v8f acc[4];
#pragma unroll
      for (int j = 0; j < 4; ++j) {
        float bias = b1[j * 16 + nn];
#pragma unroll
        for (int r = 0; r < 8; ++r) acc[j][r] = bias;
      }
      gemm16x64<true>(af, W1, E_, lane, acc);
#pragma unroll
      for (int j = 0; j < 4; ++j)
#pragma unroll
        for (int r = 0; r < 8; ++r) {
          int idx = (row0 + r + 8 * half) * E_ + j * 16 + nn;
          float z = acc[j][r];
          sH[idx] = silu_f(z);
          sSp[idx] = silu_df(z);
        }
    }
    __syncthreads();

    // 3) Pred = sH @ W2^T + b2 ; dp = coeff/B * (2/E) * (Pred - vals)
    {
      v2f af[16];
#pragma unroll
      for (int k = 0; k < 16; ++k) af[k] = ldA(sH, E_, row0, k * 4, lane);
      v8f acc[4];
#pragma unroll
      for (int j = 0; j < 4; ++j) {
        float bias = b2[j * 16 + nn];
#pragma unroll
        for (int r = 0; r < 8; ++r) acc[j][r] = bias;
      }
      gemm16x64<true>(af, W2, E_, lane, acc);
#pragma unroll
      for (int j = 0; j < 4; ++j)
#pragma unroll
        for (int r = 0; r < 8; ++r) {
          int trow = row0 + r + 8 * half;
          int tl = t0 + trow;  // local t within split
          float val =
              valA[(size_t)(b * T_ + tbase + tl) * E_ + j * 16 + nn] * invV[j * 16 + nn];
          sDp[trow * E_ + j * 16 + nn] =
              coeffS[tl] * (1.f / B_) * (2.f / E_) * (acc[j][r] - val);
        }
    }
    __syncthreads();

    // 4) dh = sDp @ W2 ; dhpre = dh * silu'(hpre)  (overwrite sSp in place)
    {
      v2f af[16];
#pragma unroll
      for (int k = 0; k < 16; ++k) af[k] = ldA(sDp, E_, row0, k * 4, lane);
      v8f acc[4] = {};
      gemm16x64<false>(af, W2, E_, lane, acc);
#pragma unroll
      for (int j = 0; j < 4; ++j)
#pragma unroll
        for (int r = 0; r < 8; ++r) {
          int idx = (row0 + r + 8 * half) * E_ + j * 16 + nn;
          sSp[idx] = acc[j][r] * sSp[idx];  // lane-private RMW, same owner as step 2
        }
    }
    __syncthreads();

    // 5) accumulate W1p += dhpre^T @ keys ; W2p += dp^T @ H ; bias column sums
#pragma unroll 4
    for (int k0 = 0; k0 < TM_; k0 += 4) {
      v2f aw1 = ldAt(sSp, E_, m0, k0, lane);
      w1a0 = wmma4(aw1, ldBn(sK, E_, n0a,      k0, lane), w1a0);
      w1a1 = wmma4(aw1, ldBn(sK, E_, n0a + 16, k0, lane), w1a1);
      v2f aw2 = ldAt(sDp, E_, m0, k0, lane);
      w2a0 = wmma4(aw2, ldBn(sH, E_, n0a,      k0, lane), w2a0);
      w2a1 = wmma4(aw2, ldBn(sH, E_, n0a + 16, k0, lane), w2a1);
    }
    if (tid < E_) {
      for (int t = 0; t < TM_; ++t) rb += sSp[t * E_ + tid];           // b1 grads
    } else if (tid < 2 * E_) {
      int c = tid - E_;
      for (int t = 0; t < TM_; ++t) rb += sDp[t * E_ + c];             // b2 grads
    }
    __syncthreads();
  }

  // write per-split partial accumulations
  const size_t pbase = (size_t)(b * NSPLIT_ + split) * E_ * E_;
  float* W1o = W1p + pbase;
  float* W2o = W2p + pbase;
#pragma unroll
  for (int r = 0; r < 8; ++r) {
    int mr = m0 + r + 8 * half;
    W1o[mr * E_ + n0a + nn]      = w1a0[r];
    W1o[mr * E_ + n0a + 16 + nn] = w1a1[r];
    W2o[mr * E_ + n0a + nn]      = w2a0[r];
    W2o[mr * E_ + n0a + 16 + nn] = w2a1[r];
  }
  if (tid < E_)          b1p[(b * NSPLIT_ + split) * E_ + tid]      = rb;
  else if (tid < 2 * E_) b2p[(b * NSPLIT_ + split) * E_ + tid - E_] = rb;
}

// ------- kernel 3b: fold partials -> final fast weights (deterministic order) -------

__global__ void nm_fwreduce(const float* __restrict__ W1, const float* __restrict__ b1,
                            const float* __restrict__ W2, const float* __restrict__ b2,
                            const float* __restrict__ W1p, const float* __restrict__ W2p,
                            const float* __restrict__ b1p, const float* __restrict__ b2p,
                            float* __restrict__ W1f, float* __restrict__ W2f,
                            float* __restrict__ b1f, float* __restrict__ b2f) {
  const int b = blockIdx.x;
  const int tid = threadIdx.x;
  const float decay = __powf(ALPHA_, (float)T_);
  for (int i = tid; i < E_ * E_; i += blockDim.x) {
    float s1 = decay * W1[i], s2 = decay * W2[i];
#pragma unroll
    for (int sp = 0; sp < NSPLIT_; ++sp) {
      s1 += W1p[(size_t)(b * NSPLIT_ + sp) * E_ * E_ + i];
      s2 += W2p[(size_t)(b * NSPLIT_ + sp) * E_ * E_ + i];
    }
    W1f[(size_t)b * E_ * E_ + i] = s1;
    W2f[(size_t)b * E_ * E_ + i] = s2;
  }
  if (tid < E_) {
    float s1 = decay * b1[tid], s2 = decay * b2[tid];
#pragma unroll
    for (int sp = 0; sp < NSPLIT_; ++sp) {
      s1 += b1p[(b * NSPLIT_ + sp) * E_ + tid];
      s2 += b2p[(b * NSPLIT_ + sp) * E_ + tid];
    }
    b1f[b * E_ + tid] = s1;
    b2f[b * E_ + tid] = s2;
  }
}

// ------------- kernel 4: retrieval through per-batch fast weights -------------

__global__ void nm_retrieve(const float* __restrict__ qA,
                            const float* __restrict__ norms,
                            const float* __restrict__ W1f, const float* __restrict__ W2f,
                            const float* __restrict__ b1f, const float* __restrict__ b2f,
                            float* __restrict__ out) {
  extern __shared__ float sm2[];
  float* sQ  = sm2;             // TM_*E_
  float* sH2 = sQ + TM_ * E_;   // TM_*E_
  __shared__ float invQ[E_];

  const int nTb = T_ / TM_;
  const int b = blockIdx.x / nTb;
  const int t0 = (blockIdx.x % nTb) * TM_;
  const int tid = threadIdx.x, wave = tid >> 5, lane = tid & 31;
  const int row0 = wave * 16, half = lane >> 4, nn = lane & 15;

  if (tid < E_)
    invQ[tid] = 1.f / fmaxf(sqrtf(norms[2 * B_ * E_ + b * E_ + tid]), 1e-12f);
  __syncthreads();

  const float* qsrc = qA + (size_t)(b * T_ + t0) * E_;
  for (int i = tid; i < TM_ * E_; i += blockDim.x) sQ[i] = qsrc[i] * invQ[i & (E_ - 1)];
  __syncthreads();

  const float* W1b = W1f + (size_t)b * E_ * E_;
  const float* W2b = W2f + (size_t)b * E_ * E_;

  {  // h2 = silu(sQ @ W1f^T + b1f)
    v2f af[16];
#pragma unroll
    for (int k = 0; k < 16; ++k) af[k] = ldA(sQ, E_, row0, k * 4, lane);
    v8f acc[4];
#pragma unroll
    for (int j = 0; j < 4; ++j) {
      float bias = b1f[b * E_ + j * 16 + nn];
#pragma unroll
      for (int r = 0; r < 8; ++r) acc[j][r] = bias;
    }
    gemm16x64<true>(af, W1b, E_, lane, acc);
#pragma unroll
    for (int j = 0; j < 4; ++j)
#pragma unroll
      for (int r = 0; r < 8; ++r)
        sH2[(row0 + r + 8 * half) * E_ + j * 16 + nn] = silu_f(acc[j][r]);
  }
  __syncthreads();

  {  // out = sH2 @ W2f^T + b2f
    v2f af[16];
#pragma unroll
    for (int k = 0; k < 16; ++k) af[k] = ldA(sH2, E_, row0, k * 4, lane);
    v8f acc[4];
#pragma unroll
    for (int j = 0; j < 4; ++j) {
      float bias = b2f[b * E_ + j * 16 + nn];
#pragma unroll
      for (int r = 0; r < 8; ++r) acc[j][r] = bias;
    }
    gemm16x64<true>(af, W2b, E_, lane, acc);
#pragma unroll
    for (int j = 0; j < 4; ++j)
#pragma unroll
      for (int r = 0; r < 8; ++r)
        out[(size_t)(b * T_ + t0 + row0 + r + 8 * half) * E_ + j * 16 + nn] = acc[j][r];
  }
}

// ---------------------------------- launch ----------------------------------

extern "C" void kernel_launch(void* const* d_in, const int* in_sizes, int n_in,
                              void* d_out, int out_size, void* d_ws, size_t ws_size,
                              hipStream_t stream) {
  (void)in_sizes; (void)n_in; (void)out_size; (void)ws_size;
  const float* x  = (const float*)d_in[0];
  const float* Kw = (const float*)d_in[1];
  const float* Qw = (const float*)d_in[2];
  const float* Vw = (const float*)d_in[3];
  const float* W1 = (const float*)d_in[4];
  const float* b1 = (const float*)d_in[5];
  const float* W2 = (const float*)d_in[6];
  const float* b2 = (const float*)d_in[7];
  float* out = (float*)d_out;

  float* ws = (float*)d_ws;
  const size_t BTE = (size_t)B_ * T_ * E_;
  const size_t EE  = (size_t)E_ * E_;
  float* keyA  = ws;                         // 3 contiguous [B,T,E] arrays
  float* valA  = keyA + BTE;
  float* qA    = valA + BTE;
  float* norms = qA + BTE;                   // 3*B*E sumsq (K,V,Q)
  float* W1f   = norms + 3 * B_ * E_;        // B*E*E
  float* W2f   = W1f + B_ * EE;
  float* b1f   = W2f + B_ * EE;
  float* b2f   = b1f + B_ * E_;
  float* W1p   = b2f + B_ * E_;              // B*NSPLIT partials
  float* W2p   = W1p + B_ * NSPLIT_ * EE;
  float* b1p   = W2p + B_ * NSPLIT_ * EE;
  float* b2p   = b1p + B_ * NSPLIT_ * E_;

  const int projSm = TM_ * E_ * sizeof(float);        // 32 KB
  const int updSm  = 4 * TM_ * E_ * sizeof(float);    // 128 KB (LDS up to 320 KB/WGP)
  const int retSm  = 2 * TM_ * E_ * sizeof(float);    // 64 KB

  nm_proj<<<B_ * (T_ / TM_), 256, projSm, stream>>>(x, Kw, Qw, Vw, keyA, valA, qA);
  nm_colnorm<<<(3 * B_ * E_) / 8, 256, 0, stream>>>(keyA, norms);
  nm_update<<<B_ * NSPLIT_, 256, updSm, stream>>>(keyA, valA, norms, W1, b1, W2, b2,
                                                  W1p, W2p, b1p, b2p);
  nm_fwreduce<<<B_, 256, 0, stream>>>(W1, b1, W2, b2, W1p, W2p, b1p, b2p,
                                      W1f, W2f, b1f, b2f);
  nm_retrieve<<<B_ * (T_ / TM_), 256, retSm, stream>>>(qA, norms, W1f, W2f, b1f, b2f, out);
}